// GAT_67465346285900
// MI455X (gfx1250) — compile-verified
//
#include <hip/hip_runtime.h>
#include <hip/hip_bf16.h>

typedef float v2f __attribute__((ext_vector_type(2)));
typedef float v8f __attribute__((ext_vector_type(8)));

#define N_NODES 50000
#define N_EDGES 800000
#define E_TOT   (N_EDGES + N_NODES)   // with self loops

// ---------------------------------------------------------------------------
// CSR construction (by dst), done once per launch
// ---------------------------------------------------------------------------
__global__ void build_edges_kernel(const long long* __restrict__ ei,
                                   int* __restrict__ tsrc, int* __restrict__ tdst,
                                   int* __restrict__ deg) {
  int e = blockIdx.x * 256 + threadIdx.x;
  if (e >= E_TOT) return;
  int s, d;
  if (e < N_EDGES) {
    s = (int)ei[e];             // edge_index[0][e]
    d = (int)ei[N_EDGES + e];   // edge_index[1][e]
  } else {
    s = e - N_EDGES; d = s;     // self loop
  }
  tsrc[e] = s; tdst[e] = d;
  atomicAdd(&deg[d], 1);
}

// single-block inclusive scan over 50000 degrees -> rowptr[0..N]
__global__ void scan_kernel(const int* __restrict__ deg, int* __restrict__ rowptr) {
  __shared__ int buf[1024];
  __shared__ int carry;
  int t = threadIdx.x;
  if (t == 0) { carry = 0; rowptr[0] = 0; }
  __syncthreads();
  for (int base = 0; base < N_NODES; base += 1024) {
    int i = base + t;
    int v = (i < N_NODES) ? deg[i] : 0;
    buf[t] = v;
    __syncthreads();
    for (int off = 1; off < 1024; off <<= 1) {
      int add = (t >= off) ? buf[t - off] : 0;
      __syncthreads();
      buf[t] += add;
      __syncthreads();
    }
    if (i < N_NODES) rowptr[i + 1] = carry + buf[t];
    __syncthreads();
    if (t == 0) carry += buf[1023];
    __syncthreads();
  }
}

__global__ void cursor_init_kernel(const int* __restrict__ rowptr, int* __restrict__ cursor) {
  int i = blockIdx.x * 256 + threadIdx.x;
  if (i < N_NODES) cursor[i] = rowptr[i];
}

__global__ void scatter_kernel(const int* __restrict__ tsrc, const int* __restrict__ tdst,
                               int* __restrict__ cursor,
                               int* __restrict__ csr_src, int* __restrict__ csr_dst) {
  int e = blockIdx.x * 256 + threadIdx.x;
  if (e >= E_TOT) return;
  int d = tdst[e];
  int pos = atomicAdd(&cursor[d], 1);
  csr_src[pos] = tsrc[e];
  csr_dst[pos] = d;
}

// ---------------------------------------------------------------------------
// GEMM  C[M,Ncols] = A[M,K] * W[Ncols,K]^T (+bias) using v_wmma_f32_16x16x4_f32
// One wave computes a 16x64 tile; software-pipelined so the 5 loads of step
// k+4 are in flight while the 4 WMMAs of step k execute.
// A frag: lane<16 holds A[m0+lane, k..k+1], lane>=16 holds A[m0+lane-16, k+2..k+3]
// B frag (B=W^T, B[k,n]=W[n,k]): lane<16 holds W[n0+lane, k..k+1], etc. (mirror)
// C frag: vgpr r -> row m0 + r + 8*(lane>>4), col n0 + (lane&15)
// ---------------------------------------------------------------------------
__global__ void gemm_nt_wmma(const float* __restrict__ A, const float* __restrict__ W,
                             const float* __restrict__ bias, float* __restrict__ C,
                             int M, int K, int ldc) {
  int wave = threadIdx.x >> 5;
  int lane = threadIdx.x & 31;
  int m0 = (blockIdx.x * 4 + wave) * 16;
  if (m0 >= M) return;
  int n0 = blockIdx.y * 64;
  int row  = lane & 15;
  int koff = (lane >> 4) << 1;

  const float* Ap = A + (size_t)(m0 + row) * K + koff;
  const float* Wp = W + (size_t)(n0 + row) * K + koff;

  v8f acc[4] = {};
  // prologue: fragments for k = 0
  v2f a  = *(const v2f*)(Ap);
  v2f b0 = *(const v2f*)(Wp);
  v2f b1 = *(const v2f*)(Wp + (size_t)16 * K);
  v2f b2 = *(const v2f*)(Wp + (size_t)32 * K);
  v2f b3 = *(const v2f*)(Wp + (size_t)48 * K);
  for (int k = 0; k < K; k += 4) {
    int kn = (k + 4 < K) ? (k + 4) : 0;   // tail reloads k=0 (in-bounds, unused)
    v2f an  = *(const v2f*)(Ap + kn);
    v2f bn0 = *(const v2f*)(Wp + kn);
    v2f bn1 = *(const v2f*)(Wp + (size_t)16 * K + kn);
    v2f bn2 = *(const v2f*)(Wp + (size_t)32 * K + kn);
    v2f bn3 = *(const v2f*)(Wp + (size_t)48 * K + kn);
    acc[0] = __builtin_amdgcn_wmma_f32_16x16x4_f32(false, a, false, b0, (short)0, acc[0], false, false);
    acc[1] = __builtin_amdgcn_wmma_f32_16x16x4_f32(false, a, false, b1, (short)0, acc[1], false, false);
    acc[2] = __builtin_amdgcn_wmma_f32_16x16x4_f32(false, a, false, b2, (short)0, acc[2], false, false);
    acc[3] = __builtin_amdgcn_wmma_f32_16x16x4_f32(false, a, false, b3, (short)0, acc[3], false, false);
    a = an; b0 = bn0; b1 = bn1; b2 = bn2; b3 = bn3;
  }

  int crow = m0 + ((lane >> 4) << 3);
  int ccol = n0 + (lane & 15);
#pragma unroll
  for (int j = 0; j < 4; ++j) {
    int col = ccol + j * 16;
    float bv = bias ? bias[col] : 0.0f;
#pragma unroll
    for (int r = 0; r < 8; ++r)
      C[(size_t)(crow + r) * ldc + col] = acc[j][r] + bv;
  }
}

// ---------------------------------------------------------------------------
// alpha_s[n,h] = <xh[n,h,:], a_src[h,:]> ; alpha_d likewise. One thread per (n,h).
// ---------------------------------------------------------------------------
__global__ void compute_alpha_kernel(const float* __restrict__ xh,
                                     const float* __restrict__ a_src,
                                     const float* __restrict__ a_dst,
                                     float* __restrict__ alpha_s,
                                     float* __restrict__ alpha_d) {
  int idx = blockIdx.x * 256 + threadIdx.x;      // n*4 + h
  if (idx >= N_NODES * 4) return;
  int h = idx & 3;
  const float* row = xh + (size_t)idx * 64;
  const float* as  = a_src + h * 64;
  const float* ad  = a_dst + h * 64;
  float s1 = 0.f, s2 = 0.f;
  for (int f = 0; f < 64; f += 4) {
    float4 xv = *(const float4*)(row + f);
    float4 av = *(const float4*)(as + f);
    float4 dv = *(const float4*)(ad + f);
    s1 += xv.x * av.x + xv.y * av.y + xv.z * av.z + xv.w * av.w;
    s2 += xv.x * dv.x + xv.y * dv.y + xv.z * dv.z + xv.w * dv.w;
  }
  alpha_s[idx] = s1;
  alpha_d[idx] = s2;
}

// ---------------------------------------------------------------------------
// Per-node softmax stats (max + sum), wave per node, (8 edges x 4 heads) lanes.
// ---------------------------------------------------------------------------
__global__ void node_stats_kernel(const int* __restrict__ rowptr, const int* __restrict__ csr_src,
                                  const float* __restrict__ alpha_s, const float* __restrict__ alpha_d,
                                  float* __restrict__ mbuf, float* __restrict__ sbuf) {
  int wave = threadIdx.x >> 5, lane = threadIdx.x & 31;
  int n = blockIdx.x * 8 + wave;
  if (n >= N_NODES) return;
  int h = lane & 3, j = lane >> 2;
  int start = rowptr[n], end = rowptr[n + 1];
  float ad = alpha_d[n * 4 + h];

  float lmax = -3.4e38f;
  for (int p = start + j; p < end; p += 8) {
    float v = alpha_s[csr_src[p] * 4 + h] + ad;
    v = v > 0.f ? v : 0.2f * v;                 // leaky_relu 0.2
    lmax = fmaxf(lmax, v);
  }
  lmax = fmaxf(lmax, __shfl_xor(lmax, 4, 32));
  lmax = fmaxf(lmax, __shfl_xor(lmax, 8, 32));
  lmax = fmaxf(lmax, __shfl_xor(lmax, 16, 32));

  float lsum = 0.f;
  for (int p = start + j; p < end; p += 8) {
    float v = alpha_s[csr_src[p] * 4 + h] + ad;
    v = v > 0.f ? v : 0.2f * v;
    lsum += __expf(v - lmax);
  }
  lsum += __shfl_xor(lsum, 4, 32);
  lsum += __shfl_xor(lsum, 8, 32);
  lsum += __shfl_xor(lsum, 16, 32);

  if (lane < 4) { mbuf[n * 4 + h] = lmax; sbuf[n * 4 + h] = lsum; }
}

// ---------------------------------------------------------------------------
// Per-edge normalized attention weights (float4 per edge, one per head)
// ---------------------------------------------------------------------------
__global__ void edge_weights_kernel(const int* __restrict__ csr_src, const int* __restrict__ csr_dst,
                                    const float* __restrict__ alpha_s, const float* __restrict__ alpha_d,
                                    const float* __restrict__ mbuf, const float* __restrict__ sbuf,
                                    float4* __restrict__ wbuf) {
  int e = blockIdx.x * 256 + threadIdx.x;
  if (e >= E_TOT) return;
  int s = csr_src[e], d = csr_dst[e];
  float4 as = ((const float4*)alpha_s)[s];
  float4 ad = ((const float4*)alpha_d)[d];
  float4 mm = ((const float4*)mbuf)[d];
  float4 ss = ((const float4*)sbuf)[d];
  float4 w; float v;
  v = as.x + ad.x; v = v > 0.f ? v : 0.2f * v; w.x = __expf(v - mm.x) / ss.x;
  v = as.y + ad.y; v = v > 0.f ? v : 0.2f * v; w.y = __expf(v - mm.y) / ss.y;
  v = as.z + ad.z; v = v > 0.f ? v : 0.2f * v; w.z = __expf(v - mm.z) / ss.z;
  v = as.w + ad.w; v = v > 0.f ? v : 0.2f * v; w.w = __expf(v - mm.w) / ss.w;
  wbuf[e] = w;
}

// ---------------------------------------------------------------------------
// Fused: weighted aggregation + head-mean + bias + residual (+leaky_relu 0.01).
// Wave per node; lane l owns hf = l + 32*r, r=0..7 (h = r>>1, f = l + 32*(r&1)).
// Next source row is prefetched (global_prefetch_b8) one edge ahead; lanes use
// a 32B stride so the wave covers all eight 128B lines of the 1KB row.
// ---------------------------------------------------------------------------
__global__ void node_aggregate_kernel(const int* __restrict__ rowptr, const int* __restrict__ csr_src,
                                      const float4* __restrict__ wbuf, const float* __restrict__ xh,
                                      const float* __restrict__ bias, const float* __restrict__ resid,
                                      float* __restrict__ out, int final_layer) {
  int wave = threadIdx.x >> 5, lane = threadIdx.x & 31;
  int n = blockIdx.x * 8 + wave;
  if (n >= N_NODES) return;
  int start = rowptr[n], end = rowptr[n + 1];

  float a0 = 0.f, a1 = 0.f, a2 = 0.f, a3 = 0.f, a4 = 0.f, a5 = 0.f, a6 = 0.f, a7 = 0.f;
  int src = csr_src[start];                       // deg >= 1 always (self loop)
  __builtin_prefetch(xh + (size_t)src * 256 + lane * 8, 0, 1);
  for (int p = start; p < end; ++p) {
    int nsrc = (p + 1 < end) ? csr_src[p + 1] : src;
    __builtin_prefetch(xh + (size_t)nsrc * 256 + lane * 8, 0, 1);
    float4 w = wbuf[p];
    const float* r = xh + (size_t)src * 256 + lane;
    a0 += w.x * r[0];   a1 += w.x * r[32];
    a2 += w.y * r[64];  a3 += w.y * r[96];
    a4 += w.z * r[128]; a5 += w.z * r[160];
    a6 += w.w * r[192]; a7 += w.w * r[224];
    src = nsrc;
  }
  float f0 = (a0 + a2 + a4 + a6) * 0.25f;   // feature c0 = lane
  float f1 = (a1 + a3 + a5 + a7) * 0.25f;   // feature c1 = lane+32
  int c0 = lane, c1 = lane + 32;
  size_t base = (size_t)n * 64;
  float r0 = resid[base + c0], r1 = resid[base + c1];
  f0 += bias[c0] + r0;
  f1 += bias[c1] + r1;
  if (!final_layer) {
    f0 = f0 > 0.f ? f0 : 0.01f * f0;
    f1 = f1 > 0.f ? f1 : 0.01f * f1;
  }
  out[base + c0] = f0;
  out[base + c1] = f1;
}

// ---------------------------------------------------------------------------
extern "C" void kernel_launch(void* const* d_in, const int* in_sizes, int n_in,
                              void* d_out, int out_size, void* d_ws, size_t ws_size,
                              hipStream_t stream) {
  (void)in_sizes; (void)n_in; (void)out_size; (void)ws_size;
  const float*     x   = (const float*)d_in[0];
  const long long* ei  = (const long long*)d_in[1];
  const float* W1  = (const float*)d_in[2];
  const float* as1 = (const float*)d_in[3];
  const float* ad1 = (const float*)d_in[4];
  const float* b1  = (const float*)d_in[5];
  const float* W2  = (const float*)d_in[6];
  const float* as2 = (const float*)d_in[7];
  const float* ad2 = (const float*)d_in[8];
  const float* b2  = (const float*)d_in[9];
  const float* W3  = (const float*)d_in[10];
  const float* as3 = (const float*)d_in[11];
  const float* ad3 = (const float*)d_in[12];
  const float* b3  = (const float*)d_in[13];
  const float* rw1 = (const float*)d_in[14];
  const float* rb1 = (const float*)d_in[15];
  const float* rw2 = (const float*)d_in[16];
  const float* rb2 = (const float*)d_in[17];
  float* out = (float*)d_out;

  // workspace carve-up (256B aligned)
  char* ws = (char*)d_ws;
  size_t off = 0;
  auto carve = [&](size_t bytes) -> void* {
    void* p = ws + off;
    off = (off + bytes + 255) & ~(size_t)255;
    return p;
  };
  int*    tsrc    = (int*)carve((size_t)E_TOT * 4);
  int*    tdst    = (int*)carve((size_t)E_TOT * 4);
  int*    deg     = (int*)carve((size_t)N_NODES * 4);
  int*    rowptr  = (int*)carve((size_t)(N_NODES + 1) * 4);
  int*    cursor  = (int*)carve((size_t)N_NODES * 4);
  int*    csr_src = (int*)carve((size_t)E_TOT * 4);
  int*    csr_dst = (int*)carve((size_t)E_TOT * 4);
  float*  xh      = (float*)carve((size_t)N_NODES * 256 * 4);
  float*  alpha_s = (float*)carve((size_t)N_NODES * 4 * 4);
  float*  alpha_d = (float*)carve((size_t)N_NODES * 4 * 4);
  float*  mbuf    = (float*)carve((size_t)N_NODES * 4 * 4);
  float*  sbuf    = (float*)carve((size_t)N_NODES * 4 * 4);
  float4* wbuf    = (float4*)carve((size_t)E_TOT * 16);
  float*  res1    = (float*)carve((size_t)N_NODES * 64 * 4);
  float*  hbuf    = (float*)carve((size_t)N_NODES * 64 * 4);
  // x_res2 lives directly in d_out (read-modify-write by final aggregate)

  dim3 blk256(256);
  dim3 egrid((E_TOT + 255) / 256);
  dim3 ngrid((N_NODES + 255) / 256);
  dim3 nhgrid((N_NODES * 4 + 255) / 256);
  dim3 wgrid((N_NODES + 7) / 8);          // wave-per-node kernels, 8 waves/block
  dim3 gblock(128);                        // 4 waves
  dim3 grid_res((N_NODES / 16 + 3) / 4, 1);
  dim3 grid_xh((N_NODES / 16 + 3) / 4, 4);

  // ---- CSR build (indices reused by all 3 layers) ----
  hipMemsetAsync(deg, 0, (size_t)N_NODES * 4, stream);
  build_edges_kernel<<<egrid, blk256, 0, stream>>>(ei, tsrc, tdst, deg);
  scan_kernel<<<1, 1024, 0, stream>>>(deg, rowptr);
  cursor_init_kernel<<<ngrid, blk256, 0, stream>>>(rowptr, cursor);
  scatter_kernel<<<egrid, blk256, 0, stream>>>(tsrc, tdst, cursor, csr_src, csr_dst);

  // ---- residual projections (WMMA f32) ----
  gemm_nt_wmma<<<grid_res, gblock, 0, stream>>>(x, rw1, rb1, res1, N_NODES, 128, 64);
  gemm_nt_wmma<<<grid_res, gblock, 0, stream>>>(x, rw2, rb2, out,  N_NODES, 128, 64);

  // ---- layer 1 ----
  gemm_nt_wmma<<<grid_xh, gblock, 0, stream>>>(x, W1, nullptr, xh, N_NODES, 128, 256);
  compute_alpha_kernel<<<nhgrid, blk256, 0, stream>>>(xh, as1, ad1, alpha_s, alpha_d);
  node_stats_kernel<<<wgrid, blk256, 0, stream>>>(rowptr, csr_src, alpha_s, alpha_d, mbuf, sbuf);
  edge_weights_kernel<<<egrid, blk256, 0, stream>>>(csr_src, csr_dst, alpha_s, alpha_d, mbuf, sbuf, wbuf);
  node_aggregate_kernel<<<wgrid, blk256, 0, stream>>>(rowptr, csr_src, wbuf, xh, b1, res1, hbuf, 0);

  // ---- layer 2 ----
  gemm_nt_wmma<<<grid_xh, gblock, 0, stream>>>(hbuf, W2, nullptr, xh, N_NODES, 64, 256);
  compute_alpha_kernel<<<nhgrid, blk256, 0, stream>>>(xh, as2, ad2, alpha_s, alpha_d);
  node_stats_kernel<<<wgrid, blk256, 0, stream>>>(rowptr, csr_src, alpha_s, alpha_d, mbuf, sbuf);
  edge_weights_kernel<<<egrid, blk256, 0, stream>>>(csr_src, csr_dst, alpha_s, alpha_d, mbuf, sbuf, wbuf);
  node_aggregate_kernel<<<wgrid, blk256, 0, stream>>>(rowptr, csr_src, wbuf, xh, b2, res1, hbuf, 0);

  // ---- layer 3 (residual = x_res2 already in d_out; final write to d_out) ----
  gemm_nt_wmma<<<grid_xh, gblock, 0, stream>>>(hbuf, W3, nullptr, xh, N_NODES, 64, 256);
  compute_alpha_kernel<<<nhgrid, blk256, 0, stream>>>(xh, as3, ad3, alpha_s, alpha_d);
  node_stats_kernel<<<wgrid, blk256, 0, stream>>>(rowptr, csr_src, alpha_s, alpha_d, mbuf, sbuf);
  edge_weights_kernel<<<egrid, blk256, 0, stream>>>(csr_src, csr_dst, alpha_s, alpha_d, mbuf, sbuf, wbuf);
  node_aggregate_kernel<<<wgrid, blk256, 0, stream>>>(rowptr, csr_src, wbuf, xh, b3, out, out, 1);
}